// NTM_27152783245908
// MI455X (gfx1250) — compile-verified
//
#include <hip/hip_runtime.h>
#include <hip/hip_bf16.h>
#include <math.h>

// ---------------------------------------------------------------------------
// NTM step for MI455X (gfx1250, wave32).
// Bandwidth-bound fp32 workload (~390MB @ 23.3TB/s ~= 17us).
// WMMA F32_16X16X4_F32 used for the 8192x4096 x 4096x2 similarity GEMM.
// ---------------------------------------------------------------------------

typedef __attribute__((ext_vector_type(2))) float v2f;
typedef __attribute__((ext_vector_type(8))) float v8f;

#define N_ROWS 8192
#define W_DIM  4096
#define XI_DIM 16396
#define OUT_DIM 325

// ws layout (float offsets)
#define WS_H2     0        // 72
#define WS_PRM    128      // 32: [0]=g_r [1..3]=s_r [4]=gamma_r [5]=beta_r [6]=normk_r ; +8 same for write head
#define WS_KR     160      // 4096
#define WS_KW     4256     // 4096
#define WS_ER     8352     // 4096
#define WS_AD     12448    // 4096
#define WS_NUMR   16544    // 8192
#define WS_NUMW   24736    // 8192
#define WS_RNORM  32928    // 8192
#define WS_RPART  41120    // 128*4096
#define WS_READ   565408   // 4096
#define WS_LOGITS 569504   // 325

// d_out layout (float offsets): out(325) | rw(8192) | ww(8192) | mem_new(8192*4096)
#define DO_OUT 0
#define DO_RW  325
#define DO_WW  8517
#define DO_MEM 16709

__device__ __forceinline__ float sigmoidf_(float x) { return 1.f / (1.f + expf(-x)); }
__device__ __forceinline__ float softplusf_(float x) { return (x > 20.f) ? x : log1pf(expf(x)); }

// ---------------------------------------------------------------- K1: MLP ---
__global__ __launch_bounds__(128) void k_mlp(const float* __restrict__ X,
                                             const float* __restrict__ W1, const float* __restrict__ b1,
                                             const float* __restrict__ W2, const float* __restrict__ b2,
                                             float* __restrict__ h2out) {
  __shared__ float h1[48];
  int t = threadIdx.x;
  if (t < 48) {
    float a = b1[t];
    for (int k = 0; k < 14; ++k) a += X[k] * W1[t * 14 + k];
    h1[t] = fmaxf(a, 0.f);
  }
  __syncthreads();
  if (t < 72) {
    float a = b2[t];
    for (int k = 0; k < 48; ++k) a += h1[k] * W2[t * 48 + k];
    h2out[t] = fmaxf(a, 0.f);
  }
}

// ------------------------------------------------- K2: xi GEMV + nonlinear ---
__global__ __launch_bounds__(256) void k_xi(const float* __restrict__ Wxi,
                                            const float* __restrict__ bxi,
                                            const float* __restrict__ h2,
                                            float* __restrict__ kr, float* __restrict__ kw,
                                            float* __restrict__ er, float* __restrict__ ad,
                                            float* __restrict__ prm) {
  __shared__ float hs[72];
  if (threadIdx.x < 72) hs[threadIdx.x] = h2[threadIdx.x];
  __syncthreads();
  int i = blockIdx.x * 256 + threadIdx.x;
  if (i < XI_DIM) {
    const float* w = Wxi + (size_t)i * 72;
    float acc = bxi[i];
    for (int k = 0; k < 72; ++k) acc += w[k] * hs[k];
    if (i < W_DIM + 6) {                       // read-head params
      int j = i;
      if (j < W_DIM)            kr[j] = tanhf(acc) + 1e-16f;
      else if (j == W_DIM)      prm[0] = sigmoidf_(acc);
      else if (j <= W_DIM + 3)  prm[1 + (j - W_DIM - 1)] = acc;    // raw s (softmaxed in K3)
      else if (j == W_DIM + 4)  prm[4] = 1.f + softplusf_(acc);
      else                      prm[5] = softplusf_(acc);
    } else if (i < 2 * (W_DIM + 6)) {          // write-head params
      int j = i - (W_DIM + 6);
      if (j < W_DIM)            kw[j] = tanhf(acc) + 1e-16f;
      else if (j == W_DIM)      prm[8] = sigmoidf_(acc);
      else if (j <= W_DIM + 3)  prm[9 + (j - W_DIM - 1)] = acc;
      else if (j == W_DIM + 4)  prm[12] = 1.f + softplusf_(acc);
      else                      prm[13] = softplusf_(acc);
    } else if (i < 2 * (W_DIM + 6) + W_DIM) {
      er[i - 2 * (W_DIM + 6)] = sigmoidf_(acc);
    } else {
      ad[i - 2 * (W_DIM + 6) - W_DIM] = tanhf(acc);
    }
  }
}

// ------------------------------------- K3: key norms + shift-softmax(3) -----
__global__ __launch_bounds__(1024) void k_norms(const float* __restrict__ kr,
                                                const float* __restrict__ kw,
                                                float* __restrict__ prm) {
  __shared__ float red[1024];
  int t = threadIdx.x;
  float ar = 0.f, aw = 0.f;
  for (int i = t; i < W_DIM; i += 1024) {
    float x = kr[i]; ar += x * x;
    float y = kw[i]; aw += y * y;
  }
  red[t] = ar; __syncthreads();
  for (int s = 512; s > 0; s >>= 1) { if (t < s) red[t] += red[t + s]; __syncthreads(); }
  if (t == 0) prm[6] = sqrtf(red[0]);
  __syncthreads();
  red[t] = aw; __syncthreads();
  for (int s = 512; s > 0; s >>= 1) { if (t < s) red[t] += red[t + s]; __syncthreads(); }
  if (t == 0) prm[14] = sqrtf(red[0]);
  if (t == 1) {
    float m = fmaxf(prm[1], fmaxf(prm[2], prm[3]));
    float e0 = expf(prm[1] - m), e1 = expf(prm[2] - m), e2 = expf(prm[3] - m);
    float s = e0 + e1 + e2;
    prm[1] = e0 / s; prm[2] = e1 / s; prm[3] = e2 / s;
  }
  if (t == 2) {
    float m = fmaxf(prm[9], fmaxf(prm[10], prm[11]));
    float e0 = expf(prm[9] - m), e1 = expf(prm[10] - m), e2 = expf(prm[11] - m);
    float s = e0 + e1 + e2;
    prm[9] = e0 / s; prm[10] = e1 / s; prm[11] = e2 / s;
  }
}

// ------------------- K4: similarity GEMM (WMMA f32 16x16x4) + row norms -----
// Per block: 16 memory rows. 8 waves split K=4096; each wave accumulates a
// 16x16 fp32 WMMA tile where B col0 = k_read, col1 = k_write (others zero).
// A fragment (16x4 f32): lane<16 holds row=lane, K pair {k,k+1};
// lane>=16 holds row=lane-16, K pair {k+2,k+3}.  B mirrors with N=lane%16.
#define TILE_K 512
#define PITCH  516   // +4 floats: rows land on distinct bank groups

__global__ __launch_bounds__(256) void k_sim(const float* __restrict__ mem,
                                             const float* __restrict__ kr,
                                             const float* __restrict__ kw,
                                             float* __restrict__ numR,
                                             float* __restrict__ numW,
                                             float* __restrict__ rnorm) {
  __shared__ float tile[16 * PITCH];
  __shared__ float cpart[8 * 32];
  __shared__ float npart[8 * 32];
  const int rowBase = blockIdx.x * 16;
  const int t = threadIdx.x;
  const int wave = t >> 5, lane = t & 31;
  const int halfsel = lane >> 4;      // 0: K{0,1}, 1: K{2,3}
  const int lrow = lane & 15;         // A: row;  B/D: column N

  v8f c = {};
  float nacc = 0.f;

  for (int k0 = 0; k0 < W_DIM; k0 += TILE_K) {
    // coalesced cooperative load of 16 x TILE_K fp32 tile
    for (int it = 0; it < (16 * TILE_K / 4) / 256; ++it) {
      int idx = it * 256 + t;              // float4 index
      int r  = idx >> 7;                   // TILE_K/4 = 128 float4 per row
      int c4 = idx & 127;
      float4 v = *(const float4*)(mem + (size_t)(rowBase + r) * W_DIM + k0 + c4 * 4);
      *(float4*)(tile + r * PITCH + c4 * 4) = v;
    }
    if (t < 16 && k0 + TILE_K < W_DIM)
      __builtin_prefetch(mem + (size_t)(rowBase + t) * W_DIM + k0 + TILE_K, 0, 1);
    __syncthreads();

    const int kbase = wave * (TILE_K / 8);   // 64 K per wave
    for (int kk = 0; kk < TILE_K / 8; kk += 4) {
      int kc = kbase + kk + 2 * halfsel;
      v2f a;
      a.x = tile[lrow * PITCH + kc];
      a.y = tile[lrow * PITCH + kc + 1];
      v2f b; b.x = 0.f; b.y = 0.f;
      int kg = k0 + kc;
      if (lrow == 0)      { b.x = kr[kg]; b.y = kr[kg + 1]; }
      else if (lrow == 1) { b.x = kw[kg]; b.y = kw[kg + 1]; }
      c = __builtin_amdgcn_wmma_f32_16x16x4_f32(false, a, false, b,
                                                (short)0, c, false, false);
      nacc += a.x * a.x + a.y * a.y;
    }
    __syncthreads();
  }

  // D tile layout: VGPR r -> M=r (lanes 0-15, N=lane) / M=8+r (lanes 16-31)
  if (lrow < 2) {
    int mofs = halfsel * 8;
    for (int r = 0; r < 8; ++r)
      cpart[wave * 32 + (mofs + r) * 2 + lrow] = c[r];
  }
  npart[wave * 32 + lane] = nacc;
  __syncthreads();

  if (t < 16) {
    float nr = 0.f, nw = 0.f, nn = 0.f;
    for (int w = 0; w < 8; ++w) {
      nr += cpart[w * 32 + t * 2 + 0];
      nw += cpart[w * 32 + t * 2 + 1];
      nn += npart[w * 32 + t] + npart[w * 32 + 16 + t];
    }
    numR[rowBase + t] = nr;
    numW[rowBase + t] = nw;
    rnorm[rowBase + t] = sqrtf(nn);
  }
}

// ----------------- K5: addressing (softmax/interp/shift/sharpen/normalize) --
__global__ __launch_bounds__(1024) void k_addr(const float* __restrict__ num,
                                               const float* __restrict__ rnorm,
                                               const float* __restrict__ prm,
                                               const float* __restrict__ wprev,
                                               float* __restrict__ wout) {
  __shared__ float wg[N_ROWS];
  __shared__ float red[1024];
  const int t = threadIdx.x;
  const float g = prm[0], s0 = prm[1], s1 = prm[2], s2 = prm[3];
  const float gamma = prm[4], beta = prm[5];
  const float nk = fmaxf(prm[6], 1e-8f);

  float sc[8];
  float lmax = -INFINITY;
  for (int j = 0; j < 8; ++j) {
    int i = t + j * 1024;
    float den = fmaxf(rnorm[i], 1e-8f) * nk;
    float v = num[i] / den * beta;
    sc[j] = v;
    lmax = fmaxf(lmax, v);
  }
  red[t] = lmax; __syncthreads();
  for (int s = 512; s > 0; s >>= 1) { if (t < s) red[t] = fmaxf(red[t], red[t + s]); __syncthreads(); }
  float m = red[0]; __syncthreads();

  float lsum = 0.f;
  for (int j = 0; j < 8; ++j) { sc[j] = expf(sc[j] - m); lsum += sc[j]; }
  red[t] = lsum; __syncthreads();
  for (int s = 512; s > 0; s >>= 1) { if (t < s) red[t] += red[t + s]; __syncthreads(); }
  float inv = 1.f / red[0]; __syncthreads();

  for (int j = 0; j < 8; ++j) {
    int i = t + j * 1024;
    wg[i] = g * (sc[j] * inv) + (1.f - g) * wprev[i];
  }
  __syncthreads();

  float wp[8]; float psum = 0.f;
  for (int j = 0; j < 8; ++j) {
    int i = t + j * 1024;
    int im = (i == 0) ? (N_ROWS - 1) : i - 1;
    int ip = (i == N_ROWS - 1) ? 0 : i + 1;
    float ws_ = s0 * wg[im] + s1 * wg[i] + s2 * wg[ip];
    float v = powf(ws_, gamma);
    wp[j] = v; psum += v;
  }
  red[t] = psum; __syncthreads();
  for (int s = 512; s > 0; s >>= 1) { if (t < s) red[t] += red[t + s]; __syncthreads(); }
  float tinv = 1.f / (red[0] + 1e-16f);
  for (int j = 0; j < 8; ++j) wout[t + j * 1024] = wp[j] * tinv;
}

// -------------- K6: memory update + partial read reduction (deterministic) --
#define RPB 64
__global__ __launch_bounds__(256) void k_update(const float* __restrict__ mem,
                                                const float* __restrict__ rw,
                                                const float* __restrict__ ww,
                                                const float* __restrict__ er,
                                                const float* __restrict__ ad,
                                                float* __restrict__ memnew,
                                                float* __restrict__ rpart) {
  const int rg = blockIdx.x;                    // 0..127
  const int col = blockIdx.y * 1024 + threadIdx.x * 4;
  const float4 e4 = *(const float4*)(er + col);
  const float4 a4 = *(const float4*)(ad + col);
  float ax = 0.f, ay = 0.f, az = 0.f, aw = 0.f;
  const int rowBase = rg * RPB;
  for (int r = 0; r < RPB; ++r) {
    const int row = rowBase + r;
    const float wwi = ww[row], rwi = rw[row];
    const float4 m4 = *(const float4*)(mem + (size_t)row * W_DIM + col);
    float ox = m4.x * (1.f - wwi * e4.x) + wwi * a4.x;
    float oy = m4.y * (1.f - wwi * e4.y) + wwi * a4.y;
    float oz = m4.z * (1.f - wwi * e4.z) + wwi * a4.z;
    float ow = m4.w * (1.f - wwi * e4.w) + wwi * a4.w;
    float* o = memnew + (size_t)row * W_DIM + col;   // base misaligned for b128
    o[0] = ox; o[1] = oy; o[2] = oz; o[3] = ow;
    ax += rwi * ox; ay += rwi * oy; az += rwi * oz; aw += rwi * ow;
  }
  float* rp = rpart + (size_t)rg * W_DIM + col;
  rp[0] = ax; rp[1] = ay; rp[2] = az; rp[3] = aw;
}

__global__ __launch_bounds__(256) void k_redread(const float* __restrict__ rpart,
                                                 float* __restrict__ readv) {
  int j = blockIdx.x * 256 + threadIdx.x;
  float a = 0.f;
  for (int rg = 0; rg < 128; ++rg) a += rpart[(size_t)rg * W_DIM + j];
  readv[j] = a;
}

// ------------------------------------------ K8/K9: output GEMV + softmax ----
__global__ __launch_bounds__(256) void k_logits(const float* __restrict__ readv,
                                                const float* __restrict__ Wr,
                                                const float* __restrict__ br,
                                                float* __restrict__ logits) {
  __shared__ float red[256];
  const int o = blockIdx.x;
  float a = 0.f;
  for (int j = threadIdx.x; j < W_DIM; j += 256) a += readv[j] * Wr[(size_t)o * W_DIM + j];
  red[threadIdx.x] = a; __syncthreads();
  for (int s = 128; s > 0; s >>= 1) { if (threadIdx.x < s) red[threadIdx.x] += red[threadIdx.x + s]; __syncthreads(); }
  if (threadIdx.x == 0) logits[o] = red[0] + br[o];
}

__global__ __launch_bounds__(512) void k_softmax(const float* __restrict__ logits,
                                                 float* __restrict__ out) {
  __shared__ float red[512];
  const int t = threadIdx.x;
  float v = (t < OUT_DIM) ? logits[t] : -INFINITY;
  red[t] = v; __syncthreads();
  for (int s = 256; s > 0; s >>= 1) { if (t < s) red[t] = fmaxf(red[t], red[t + s]); __syncthreads(); }
  float m = red[0]; __syncthreads();
  float e = (t < OUT_DIM) ? expf(v - m) : 0.f;
  red[t] = e; __syncthreads();
  for (int s = 256; s > 0; s >>= 1) { if (t < s) red[t] += red[t + s]; __syncthreads(); }
  if (t < OUT_DIM) out[t] = e / red[0];
}

// ---------------------------------------------------------------------------
extern "C" void kernel_launch(void* const* d_in, const int* in_sizes, int n_in,
                              void* d_out, int out_size, void* d_ws, size_t ws_size,
                              hipStream_t stream) {
  const float* X   = (const float*)d_in[0];
  const float* rwp = (const float*)d_in[1];   // read_weights (prev)
  const float* wwp = (const float*)d_in[2];   // write_weights (prev)
  const float* mem = (const float*)d_in[3];
  const float* W1  = (const float*)d_in[4];
  const float* b1  = (const float*)d_in[5];
  const float* W2  = (const float*)d_in[6];
  const float* b2  = (const float*)d_in[7];
  const float* Wxi = (const float*)d_in[8];
  const float* bxi = (const float*)d_in[9];
  const float* Wr  = (const float*)d_in[10];
  const float* br  = (const float*)d_in[11];

  float* ws  = (float*)d_ws;
  float* out = (float*)d_out;

  k_mlp<<<1, 128, 0, stream>>>(X, W1, b1, W2, b2, ws + WS_H2);
  k_xi<<<(XI_DIM + 255) / 256, 256, 0, stream>>>(Wxi, bxi, ws + WS_H2,
      ws + WS_KR, ws + WS_KW, ws + WS_ER, ws + WS_AD, ws + WS_PRM);
  k_norms<<<1, 1024, 0, stream>>>(ws + WS_KR, ws + WS_KW, ws + WS_PRM);
  k_sim<<<N_ROWS / 16, 256, 0, stream>>>(mem, ws + WS_KR, ws + WS_KW,
      ws + WS_NUMR, ws + WS_NUMW, ws + WS_RNORM);
  k_addr<<<1, 1024, 0, stream>>>(ws + WS_NUMR, ws + WS_RNORM, ws + WS_PRM,
      rwp, out + DO_RW);
  k_addr<<<1, 1024, 0, stream>>>(ws + WS_NUMW, ws + WS_RNORM, ws + WS_PRM + 8,
      wwp, out + DO_WW);
  dim3 ugrid(N_ROWS / RPB, W_DIM / 1024);
  k_update<<<ugrid, 256, 0, stream>>>(mem, out + DO_RW, out + DO_WW,
      ws + WS_ER, ws + WS_AD, out + DO_MEM, ws + WS_RPART);
  k_redread<<<W_DIM / 256, 256, 0, stream>>>(ws + WS_RPART, ws + WS_READ);
  k_logits<<<OUT_DIM, 256, 0, stream>>>(ws + WS_READ, Wr, br, ws + WS_LOGITS);
  k_softmax<<<1, 512, 0, stream>>>(ws + WS_LOGITS, out + DO_OUT);
}